// SMoE_Momentum_11063835755041
// MI455X (gfx1250) — compile-verified
//
#include <hip/hip_runtime.h>
#include <math.h>

#define N_ROWS 16384
#define DIM    4096
#define NEXP   64
#define ALPHA  1.0f
#define BETA   0.9f

#define MTILES 2                       // 16-row tiles per wave
#define NTILES 4                       // 16-expert tiles (all 64 experts)
#define WAVES  8                       // waves per block (wave32)
#define ROWS_PER_WAVE  (16 * MTILES)   // 32
#define ROWS_PER_BLOCK (WAVES * ROWS_PER_WAVE)  // 256
#define NBLOCKS (N_ROWS / ROWS_PER_BLOCK)       // 64

typedef __attribute__((ext_vector_type(2))) float v2f;
typedef __attribute__((ext_vector_type(8))) float v8f;

__global__ __launch_bounds__(WAVES * 32)
void smome_main(const float* __restrict__ inp,
                const float* __restrict__ gate_w,
                const float* __restrict__ avg_logits,
                float* __restrict__ out,
                float* __restrict__ colsum_partial)
{
    const int tid  = threadIdx.x;
    const int wave = tid >> 5;
    const int lane = tid & 31;
    const int half = lane >> 4;   // 0: lanes 0-15, 1: lanes 16-31
    const int lr   = lane & 15;

    const int wave_row_base = blockIdx.x * ROWS_PER_BLOCK + wave * ROWS_PER_WAVE;

    // A-matrix lane pointers: row = base + mt*16 + lr ; k offset 2*half (+kb)
    const float* aptr[MTILES];
#pragma unroll
    for (int mt = 0; mt < MTILES; ++mt)
        aptr[mt] = inp + (size_t)(wave_row_base + mt * 16 + lr) * DIM + 2 * half;

    // B-matrix lane pointers: expert = t*16 + lr ; gate_w is [64,4096] row-major
    const float* bptr[NTILES];
#pragma unroll
    for (int t = 0; t < NTILES; ++t)
        bptr[t] = gate_w + (size_t)(t * 16 + lr) * DIM + 2 * half;

    // Penalty per (tile, lane-expert)
    float pen[NTILES];
#pragma unroll
    for (int t = 0; t < NTILES; ++t)
        pen[t] = ALPHA * avg_logits[t * 16 + lr];

    v8f acc[MTILES][NTILES];
#pragma unroll
    for (int mt = 0; mt < MTILES; ++mt)
#pragma unroll
        for (int t = 0; t < NTILES; ++t)
            acc[mt][t] = (v8f){0.f, 0.f, 0.f, 0.f, 0.f, 0.f, 0.f, 0.f};

    // ---- K loop: stream inp once, WMMA-accumulate all 8 tiles ----
#pragma unroll 4
    for (int kb = 0; kb < DIM; kb += 4) {
        v2f a[MTILES], b[NTILES];
#pragma unroll
        for (int mt = 0; mt < MTILES; ++mt)
            a[mt] = *(const v2f*)(aptr[mt] + kb);
#pragma unroll
        for (int t = 0; t < NTILES; ++t)
            b[t] = *(const v2f*)(bptr[t] + kb);
#pragma unroll
        for (int mt = 0; mt < MTILES; ++mt)
#pragma unroll
            for (int t = 0; t < NTILES; ++t)
                acc[mt][t] = __builtin_amdgcn_wmma_f32_16x16x4_f32(
                    false, a[mt], false, b[t], (short)0, acc[mt][t],
                    false, false);
    }

    // ---- Column sums (raw logits) for the momentum buffer ----
    // acc[mt][t][r] = logits[base + mt*16 + r + 8*half][t*16 + lr]
    float cs[NTILES];
#pragma unroll
    for (int t = 0; t < NTILES; ++t) {
        float s = 0.f;
#pragma unroll
        for (int mt = 0; mt < MTILES; ++mt)
#pragma unroll
            for (int r = 0; r < 8; ++r)
                s += acc[mt][t][r];
        s += __shfl_xor(s, 16, 32);   // combine the two half-wave row sets
        cs[t] = s;
    }

    __shared__ float ssum[WAVES][NEXP];
    if (half == 0) {
#pragma unroll
        for (int t = 0; t < NTILES; ++t)
            ssum[wave][t * 16 + lr] = cs[t];
    }
    __syncthreads();
    if (tid < NEXP) {
        float s = 0.f;
#pragma unroll
        for (int w = 0; w < WAVES; ++w)
            s += ssum[w][tid];
        colsum_partial[(size_t)blockIdx.x * NEXP + tid] = s;   // deterministic partials
    }

    // ---- Fused top-2 + softmax per row ----
#pragma unroll
    for (int mt = 0; mt < MTILES; ++mt) {
#pragma unroll
        for (int r = 0; r < 8; ++r) {
            // local top-2 among this lane's 4 candidates (one per expert tile)
            float b1 = -INFINITY, b2 = -INFINITY;
            int   i1 = 0x7fffffff, i2 = 0x7fffffff;
#pragma unroll
            for (int t = 0; t < NTILES; ++t) {
                float v = acc[mt][t][r] - pen[t];
                int   e = t * 16 + lr;
                if (v > b1 || (v == b1 && e < i1)) {
                    b2 = b1; i2 = i1; b1 = v; i1 = e;
                } else if (v > b2 || (v == b2 && e < i2)) {
                    b2 = v; i2 = e;
                }
            }
            // butterfly merge across the 16 lanes of this half-wave
#pragma unroll
            for (int m = 1; m < 16; m <<= 1) {
                float o1 = __shfl_xor(b1, m, 32);
                int  oi1 = __shfl_xor(i1, m, 32);
                float o2 = __shfl_xor(b2, m, 32);
                int  oi2 = __shfl_xor(i2, m, 32);
                if (o1 > b1 || (o1 == b1 && oi1 < i1)) {
                    if (b1 > o2 || (b1 == o2 && i1 < oi2)) { b2 = b1; i2 = i1; }
                    else                                   { b2 = o2; i2 = oi2; }
                    b1 = o1; i1 = oi1;
                } else if (o1 > b2 || (o1 == b2 && oi1 < i2)) {
                    b2 = o1; i2 = oi1;
                }
            }
            if (lr == r) {
                const int row = wave_row_base + mt * 16 + 8 * half + r;
                const float e2 = __expf(b2 - b1);       // <= 1, stable
                const float s1 = 1.0f / (1.0f + e2);
                const float s2 = e2 * s1;
                out[2 * row + 0]              = (float)i1;
                out[2 * row + 1]              = (float)i2;
                out[2 * N_ROWS + 2 * row + 0] = s1;
                out[2 * N_ROWS + 2 * row + 1] = s2;
            }
        }
    }
}

__global__ __launch_bounds__(NEXP)
void smome_finalize(const float* __restrict__ avg_logits,
                    const float* __restrict__ colsum_partial,
                    float* __restrict__ out)
{
    const int e = threadIdx.x;
    float s = 0.f;
    for (int b = 0; b < NBLOCKS; ++b)
        s += colsum_partial[(size_t)b * NEXP + e];
    const float mean = s * (1.0f / (float)N_ROWS);
    out[4 * N_ROWS + e] = BETA * avg_logits[e] + (1.0f - BETA) * mean;
}

extern "C" void kernel_launch(void* const* d_in, const int* in_sizes, int n_in,
                              void* d_out, int out_size, void* d_ws, size_t ws_size,
                              hipStream_t stream)
{
    const float* inp        = (const float*)d_in[0];
    const float* gate_w     = (const float*)d_in[1];
    const float* avg_logits = (const float*)d_in[2];
    float* out  = (float*)d_out;
    float* part = (float*)d_ws;   // NBLOCKS*NEXP = 4096 floats of scratch

    smome_main<<<NBLOCKS, WAVES * 32, 0, stream>>>(inp, gate_w, avg_logits, out, part);
    smome_finalize<<<1, NEXP, 0, stream>>>(avg_logits, part, out);
}